// GCNII_82231443849286
// MI455X (gfx1250) — compile-verified
//
#include <hip/hip_runtime.h>
#include <hip/hip_bf16.h>
#include <math.h>

// ---------------- problem constants (match reference) ----------------
#define IN_C   128
#define HID    64
#define OUT_C  64
#define NLAYER 8
#define ALPHA  0.1f
#define THETA  0.5f

typedef __attribute__((ext_vector_type(2))) float v2f;
typedef __attribute__((ext_vector_type(8))) float v8f;

__device__ __forceinline__ void atomAddF(float* p, float v) {
    __hip_atomic_fetch_add(p, v, __ATOMIC_RELAXED, __HIP_MEMORY_SCOPE_AGENT);
}

// ---------------- elementwise helpers ----------------
__global__ void fill_kernel(float* __restrict__ p, float v, int n) {
    int i = blockIdx.x * blockDim.x + threadIdx.x;
    if (i < n) p[i] = v;
}

__global__ void deg_accum_kernel(const int* __restrict__ col, float* __restrict__ deg, int E) {
    int e = blockIdx.x * blockDim.x + threadIdx.x;
    if (e < E) atomAddF(&deg[col[e]], 1.0f);
}

__global__ void rsqrt_kernel(float* __restrict__ d, int n) {
    int i = blockIdx.x * blockDim.x + threadIdx.x;
    if (i < n) d[i] = rsqrtf(d[i]);   // deg >= 1 always (self loop)
}

// normS[e] = (1-ALPHA) * dinv[row]*dinv[col]; edges e>=E are self loops (node e-E)
__global__ void norm_kernel(const int* __restrict__ row, const int* __restrict__ col,
                            const float* __restrict__ dinv, float* __restrict__ normS,
                            int E, int Etot) {
    int e = blockIdx.x * blockDim.x + threadIdx.x;
    if (e >= Etot) return;
    int r, c;
    if (e < E) { r = row[e]; c = col[e]; } else { r = c = e - E; }
    normS[e] = (1.0f - ALPHA) * dinv[r] * dinv[c];
}

// agg = ALPHA * x0
__global__ void agg_init_kernel(const float* __restrict__ x0, float* __restrict__ agg, int n) {
    int i = blockIdx.x * blockDim.x + threadIdx.x;
    if (i < n) agg[i] = ALPHA * x0[i];
}

// agg[col] += normS[e] * h[row]; 16 threads/edge, float4 per thread (64 channels)
__global__ void scatter_kernel(const float* __restrict__ h,
                               const int* __restrict__ row, const int* __restrict__ col,
                               const float* __restrict__ normS,
                               float* __restrict__ agg, int E, int Etot) {
    int t = blockIdx.x * blockDim.x + threadIdx.x;
    int e = t >> 4;
    if (e >= Etot) return;
    int cg = (t & 15) * 4;
    int r, c;
    if (e < E) { r = row[e]; c = col[e]; } else { r = c = e - E; }
    float s = normS[e];
    const float4 hv = *(const float4*)(h + (size_t)r * HID + cg);
    float* dst = agg + (size_t)c * HID + cg;
    atomAddF(dst + 0, s * hv.x);
    atomAddF(dst + 1, s * hv.y);
    atomAddF(dst + 2, s * hv.z);
    atomAddF(dst + 3, s * hv.w);
}

// ---------------- fp32 WMMA GEMM ----------------
// out[N,64] = act( scaleMM * (A[N,K] @ Wt[64,K]^T) + scaleRes*Res + bias )
// One wave per 16x16 tile; 4 waves/block cover all 64 output columns.
__global__ __launch_bounds__(128)
void gemm_wmma_kernel(const float* __restrict__ A, int K, int Nrows,
                      const float* __restrict__ Wt,
                      const float* __restrict__ bias,
                      const float* __restrict__ Res,
                      float scaleRes, float scaleMM,
                      float* __restrict__ out, float* __restrict__ out2,
                      int doRelu) {
    const int lane = threadIdx.x & 31;
    const int wave = threadIdx.x >> 5;            // 0..3 -> output col tile
    const int rowTile = blockIdx.x * 16;
    const int m    = lane & 15;
    const int half = lane >> 4;
    const int ksel = half * 2;                    // lanes 0-15: K=k,k+1 ; 16-31: K=k+2,k+3
    const int n_g  = wave * 16 + m;               // global output column

    int rowA = rowTile + m; if (rowA >= Nrows) rowA = Nrows - 1;   // safe clamp
    const float* aPtr = A  + (size_t)rowA * K + ksel;
    const float* bPtr = Wt + (size_t)n_g  * K + ksel;

    v8f c = {};
    for (int k = 0; k < K; k += 4) {
        v2f a = *(const v2f*)(aPtr + k);
        v2f b = *(const v2f*)(bPtr + k);
        // V_WMMA_F32_16X16X4_F32: D = A(16x4) * B(4x16) + C(16x16)
        c = __builtin_amdgcn_wmma_f32_16x16x4_f32(false, a, false, b,
                                                  (short)0, c, false, false);
    }

    const float bv = bias ? bias[n_g] : 0.0f;
    #pragma unroll
    for (int r = 0; r < 8; ++r) {
        int row_g = rowTile + r + 8 * half;       // C/D layout: VGPR r -> M = r (+8 for hi half)
        if (row_g >= Nrows) continue;
        size_t idx = (size_t)row_g * HID + n_g;
        float v = scaleMM * c[r] + bv;
        if (Res) v += scaleRes * Res[idx];
        if (doRelu) v = fmaxf(v, 0.0f);
        out[idx] = v;
        if (out2) out2[idx] = v;
    }
}

// ---------------- host launcher ----------------
static inline size_t roundup64(size_t x) { return (x + 63) & ~(size_t)63; }

extern "C" void kernel_launch(void* const* d_in, const int* in_sizes, int n_in,
                              void* d_out, int out_size, void* d_ws, size_t ws_size,
                              hipStream_t stream) {
    const float* x      = (const float*)d_in[0];
    const int*   ei     = (const int*)  d_in[1];
    const float* w_in   = (const float*)d_in[2];
    const float* b_in   = (const float*)d_in[3];
    const float* conv_w = (const float*)d_in[4];
    const float* w_out  = (const float*)d_in[5];
    const float* b_out  = (const float*)d_in[6];
    float* outp = (float*)d_out;

    const int N    = in_sizes[0] / IN_C;
    const int E    = in_sizes[1] / 2;
    const int Etot = E + N;
    const int* rowI = ei;
    const int* colI = ei + E;

    // workspace layout (floats)
    float* ws = (float*)d_ws;
    size_t off = 0;
    float* dinv  = ws + off; off += roundup64((size_t)N);
    float* normS = ws + off; off += roundup64((size_t)Etot);
    float* x0    = ws + off; off += roundup64((size_t)N * HID);
    float* h     = ws + off; off += roundup64((size_t)N * HID);
    float* agg   = ws + off; off += roundup64((size_t)N * HID);
    (void)ws_size;

    const int T = 256;
    const int NH = N * HID;
    const int gemmBlocks = (N + 15) / 16;

    // 1) degree (with self loop) -> dinv
    fill_kernel<<<(N + T - 1) / T, T, 0, stream>>>(dinv, 1.0f, N);
    deg_accum_kernel<<<(E + T - 1) / T, T, 0, stream>>>(colI, dinv, E);
    rsqrt_kernel<<<(N + T - 1) / T, T, 0, stream>>>(dinv, N);

    // 2) per-edge scaled norm
    norm_kernel<<<(Etot + T - 1) / T, T, 0, stream>>>(rowI, colI, dinv, normS, E, Etot);

    // 3) h = x0 = relu(x @ w_in^T + b_in)
    gemm_wmma_kernel<<<gemmBlocks, 128, 0, stream>>>(
        x, IN_C, N, w_in, b_in, nullptr, 0.0f, 1.0f, h, x0, 1);

    // 4) GCNII layers
    for (int l = 0; l < NLAYER; ++l) {
        float beta = logf(THETA / (float)(l + 1) + 1.0f);
        agg_init_kernel<<<(NH + T - 1) / T, T, 0, stream>>>(x0, agg, NH);
        {
            long long tt = (long long)Etot * 16;
            scatter_kernel<<<(int)((tt + T - 1) / T), T, 0, stream>>>(
                h, rowI, colI, normS, agg, E, Etot);
        }
        // h = relu((1-beta)*agg + beta*(agg @ W_l^T))
        gemm_wmma_kernel<<<gemmBlocks, 128, 0, stream>>>(
            agg, HID, N, conv_w + (size_t)l * HID * HID, nullptr,
            agg, 1.0f - beta, beta, h, nullptr, 1);
    }

    // 5) out = h @ w_out^T + b_out
    gemm_wmma_kernel<<<gemmBlocks, 128, 0, stream>>>(
        h, HID, N, w_out, b_out, nullptr, 0.0f, 1.0f, outp, nullptr, 0);
}